// MyModel_61933428412865
// MI455X (gfx1250) — compile-verified
//
#include <hip/hip_runtime.h>
#include <hip/hip_bf16.h>

// D = A(16x4) x B(4x16) + C with A == all-ones  =>  every row of D is the
// column-sum of B.  The matrix pipe is used as a reduction engine; the VALU
// stays free for gather address generation.
typedef __attribute__((ext_vector_type(2))) float v2f;
typedef __attribute__((ext_vector_type(8))) float v8f;

#define NCOLS 4096              // EMBED_DIM
#define NGROUPS 32              // row groups per matrix -> partial buffers
#define TOKENS_PER_GROUP 512    // 16384 tokens / 32 groups (power of two)
#define WROWS_PER_GROUP 128     // 4096 rows of W / 32 groups

// ---------------------------------------------------------------------------
// Pass A: column sums of the gathered embedding matrix (16384 x 4096 f32).
// grid = (8, NGROUPS); block = 256 threads = 8 waves (wave32).
// wave -> 64-column stripe: stripe = blockIdx.x*8 + waveId; blockIdx.y -> group.
// Software-pipelined: loads for step t+4 are in flight while step t feeds WMMA.
// ---------------------------------------------------------------------------
__global__ __launch_bounds__(256) void colsum_gather_kernel(
    const float* __restrict__ src,      // [VOCAB][NCOLS]
    const int*   __restrict__ ids,      // [16384]
    float*       __restrict__ partials) // [NGROUPS][NCOLS]
{
    __shared__ int s_ids[TOKENS_PER_GROUP];

    const int lane    = threadIdx.x & 31;
    const int wave    = threadIdx.x >> 5;
    const int group   = blockIdx.y;
    const int stripe  = blockIdx.x * 8 + wave;          // 0..63
    const int n0      = stripe * 64;                    // 64 columns per wave
    const int rowBase = group * TOKENS_PER_GROUP;

    for (int i = threadIdx.x; i < TOKENS_PER_GROUP; i += 256)
        s_ids[i] = ids[rowBase + i];
    __syncthreads();

    const v2f onesA = {1.0f, 1.0f};                     // A-matrix: all ones
    v8f acc0 = {}, acc1 = {}, acc2 = {}, acc3 = {};

    const int    halfsel = lane >> 4;                   // K-pair this lane feeds
    const int    colBase = n0 + 4 * (lane & 15);        // float4-aligned
    const float* colp    = src + colBase;

    // Prologue: issue loads for step 0.
    long long r0 = s_ids[halfsel];
    long long r1 = s_ids[2 + halfsel];
    float4 d0 = *(const float4*)(colp + r0 * (long long)NCOLS);
    float4 d1 = *(const float4*)(colp + r1 * (long long)NCOLS);

    for (int t = 0; t < TOKENS_PER_GROUP; t += 4) {
        // Branchless lookahead (wraps at group end; extra loads are harmless).
        const int tn = (t + 4) & (TOKENS_PER_GROUP - 1);
        const long long nr0 = s_ids[tn + halfsel];
        const long long nr1 = s_ids[tn + 2 + halfsel];
        const long long pr  = s_ids[((t + 8) & (TOKENS_PER_GROUP - 1)) + halfsel];
        __builtin_prefetch(colp + pr * (long long)NCOLS, 0, 1);

        const float4 nd0 = *(const float4*)(colp + nr0 * (long long)NCOLS);
        const float4 nd1 = *(const float4*)(colp + nr1 * (long long)NCOLS);

        // B operand j holds column (n0 + 4*lane + j); un-permuted at the store.
        // K-slot order inside B is irrelevant because A == 1 (sum commutes).
        v2f b0 = {d0.x, d1.x};
        v2f b1 = {d0.y, d1.y};
        v2f b2 = {d0.z, d1.z};
        v2f b3 = {d0.w, d1.w};
        acc0 = __builtin_amdgcn_wmma_f32_16x16x4_f32(false, onesA, false, b0,
                                                     (short)0, acc0, false, false);
        acc1 = __builtin_amdgcn_wmma_f32_16x16x4_f32(false, onesA, false, b1,
                                                     (short)0, acc1, false, false);
        acc2 = __builtin_amdgcn_wmma_f32_16x16x4_f32(false, onesA, false, b2,
                                                     (short)0, acc2, false, false);
        acc3 = __builtin_amdgcn_wmma_f32_16x16x4_f32(false, onesA, false, b3,
                                                     (short)0, acc3, false, false);
        d0 = nd0;
        d1 = nd1;
    }

    // Every row of D is the column sum; D vgpr0 lanes 0..15 carry N=0..15.
    if (lane < 16) {
        float* outp = partials + (size_t)group * NCOLS + n0 + 4 * lane;
        outp[0] = acc0[0];
        outp[1] = acc1[0];
        outp[2] = acc2[0];
        outp[3] = acc3[0];
    }
}

// ---------------------------------------------------------------------------
// Pass B: column sums of W (4096 x 4096 f32), linear rows, pointer marching.
// ---------------------------------------------------------------------------
__global__ __launch_bounds__(256) void colsum_dense_kernel(
    const float* __restrict__ src,      // [NCOLS][NCOLS]
    float*       __restrict__ partials) // [NGROUPS][NCOLS]
{
    const int lane   = threadIdx.x & 31;
    const int wave   = threadIdx.x >> 5;
    const int group  = blockIdx.y;
    const int stripe = blockIdx.x * 8 + wave;
    const int n0     = stripe * 64;

    const v2f onesA = {1.0f, 1.0f};
    v8f acc0 = {}, acc1 = {}, acc2 = {}, acc3 = {};

    const int halfsel = lane >> 4;
    const int colBase = n0 + 4 * (lane & 15);

    const float* p0 = src + (size_t)(group * WROWS_PER_GROUP + halfsel) * NCOLS + colBase;
    const float* p1 = p0 + 2 * (size_t)NCOLS;

    for (int t = 0; t < WROWS_PER_GROUP; t += 4) {
        const float4 d0 = *(const float4*)p0;
        const float4 d1 = *(const float4*)p1;
        p0 += 4 * (size_t)NCOLS;
        p1 += 4 * (size_t)NCOLS;

        v2f b0 = {d0.x, d1.x};
        v2f b1 = {d0.y, d1.y};
        v2f b2 = {d0.z, d1.z};
        v2f b3 = {d0.w, d1.w};
        acc0 = __builtin_amdgcn_wmma_f32_16x16x4_f32(false, onesA, false, b0,
                                                     (short)0, acc0, false, false);
        acc1 = __builtin_amdgcn_wmma_f32_16x16x4_f32(false, onesA, false, b1,
                                                     (short)0, acc1, false, false);
        acc2 = __builtin_amdgcn_wmma_f32_16x16x4_f32(false, onesA, false, b2,
                                                     (short)0, acc2, false, false);
        acc3 = __builtin_amdgcn_wmma_f32_16x16x4_f32(false, onesA, false, b3,
                                                     (short)0, acc3, false, false);
    }

    if (lane < 16) {
        float* outp = partials + (size_t)group * NCOLS + n0 + 4 * lane;
        outp[0] = acc0[0];
        outp[1] = acc1[0];
        outp[2] = acc2[0];
        outp[3] = acc3[0];
    }
}

// ---------------------------------------------------------------------------
// Pass C: result = dot(colsum_emb, colsum_W) + (BATCH*SEQ) * sum(bias).
// Single block, fixed reduction order -> bit-deterministic across replays.
// ---------------------------------------------------------------------------
__global__ __launch_bounds__(256) void final_reduce_kernel(
    const float* __restrict__ pA,   // [NGROUPS][NCOLS]
    const float* __restrict__ pB,   // [NGROUPS][NCOLS]
    const float* __restrict__ bias, // [NCOLS]
    float* __restrict__ out)
{
    __shared__ float red[256];
    const int tid = threadIdx.x;
    float acc = 0.0f, bsum = 0.0f;
    for (int d = tid; d < NCOLS; d += 256) {
        float sA = 0.0f, sB = 0.0f;
        for (int g = 0; g < NGROUPS; ++g) {
            sA += pA[g * NCOLS + d];
            sB += pB[g * NCOLS + d];
        }
        acc  += sA * sB;
        bsum += bias[d];
    }
    acc += 16384.0f * bsum;         // BATCH*SEQ = 4*4096
    red[tid] = acc;
    __syncthreads();
    for (int s = 128; s > 0; s >>= 1) {
        if (tid < s) red[tid] += red[tid + s];
        __syncthreads();
    }
    if (tid == 0) out[0] = red[0];
}

extern "C" void kernel_launch(void* const* d_in, const int* in_sizes, int n_in,
                              void* d_out, int out_size, void* d_ws, size_t ws_size,
                              hipStream_t stream) {
    const int*   ids   = (const int*)  d_in[0];   // [4, 4096] int32
    const float* embed = (const float*)d_in[1];   // [50264, 4096] f32
    const float* W     = (const float*)d_in[2];   // [4096, 4096] f32
    const float* bias  = (const float*)d_in[3];   // [4096] f32
    float* out = (float*)d_out;

    float* pA = (float*)d_ws;                     // 32*4096 f32 = 512 KB
    float* pB = pA + (size_t)NGROUPS * NCOLS;     // 32*4096 f32 = 512 KB

    dim3 grid(8, NGROUPS);
    // Pass A: gathered-embedding column sums (256 MB of HBM traffic).
    colsum_gather_kernel<<<grid, 256, 0, stream>>>(embed, ids, pA);
    // Pass B: W column sums (64 MB).
    colsum_dense_kernel<<<grid, 256, 0, stream>>>(W, pB);
    // Pass C: dot + bias -> scalar.
    final_reduce_kernel<<<1, 256, 0, stream>>>(pA, pB, bias, out);
}